// Group_81870666596633
// MI455X (gfx1250) — compile-verified
//
#include <hip/hip_runtime.h>
#include <cstdint>

// Problem constants (from reference setup_inputs)
#define BATCH 8
#define NPTS  16384
#define NGRP  1024
#define KNN_K 32

#define NEIGH_ELEMS (BATCH * NGRP * KNN_K * 3)  // 786432
#define CENTER_OFF  NEIGH_ELEMS                 // centers start here in d_out

typedef float v2f __attribute__((ext_vector_type(2)));
typedef float v8f __attribute__((ext_vector_type(8)));

// ---------------------------------------------------------------------------
// Kernel 1: Farthest Point Sampling. One workgroup per batch, 1024 threads.
// xyz for the batch lives entirely in LDS; per-thread mindist in registers.
// Writes center coords into d_out[CENTER_OFF ...].
// ---------------------------------------------------------------------------
__global__ __launch_bounds__(1024) void fps_kernel(const float* __restrict__ xyz,
                                                   float* __restrict__ out) {
    extern __shared__ char smem_raw[];
    float* sX   = (float*)smem_raw;            // NPTS
    float* sY   = sX + NPTS;                   // NPTS
    float* sZ   = sY + NPTS;                   // NPTS
    float* redV = sZ + NPTS;                   // 32
    int*   redI = (int*)(redV + 32);           // 32
    int*   sLast = redI + 32;                  // 1

    const int tid = threadIdx.x;
    const int b   = blockIdx.x;
    const float* pts = xyz + (size_t)b * NPTS * 3;
    float* ctr = out + CENTER_OFF + (size_t)b * NGRP * 3;

    // Load batch points into LDS (coalesced over interleaved layout)
    for (int i = tid; i < NPTS; i += 1024) {
        sX[i] = pts[3 * i + 0];
        sY[i] = pts[3 * i + 1];
        sZ[i] = pts[3 * i + 2];
    }
    __syncthreads();

    // Per-thread running min-distance (points tid + j*1024), in registers
    float mind[16];
#pragma unroll
    for (int j = 0; j < 16; ++j) mind[j] = 1e10f;

    if (tid == 0) {
        ctr[0] = sX[0]; ctr[1] = sY[0]; ctr[2] = sZ[0];  // first center = index 0
        *sLast = 0;
    }
    int last = 0;

    for (int it = 1; it < NGRP; ++it) {
        const float lx = sX[last], ly = sY[last], lz = sZ[last];
        float bv = -1.0f;
        int   bi = 0x7fffffff;
#pragma unroll
        for (int j = 0; j < 16; ++j) {
            const int p = tid + j * 1024;
            const float dx = sX[p] - lx;
            const float dy = sY[p] - ly;
            const float dz = sZ[p] - lz;
            const float d  = fmaf(dx, dx, fmaf(dy, dy, dz * dz));
            float m = mind[j];
            m = fminf(m, d);
            mind[j] = m;
            if (m > bv) { bv = m; bi = p; }   // strict > keeps smallest p on ties
        }
        // wave32 argmax reduce (prefer larger value; tie -> smaller index)
#pragma unroll
        for (int off = 16; off > 0; off >>= 1) {
            const float ov = __shfl_down(bv, off, 32);
            const int   oi = __shfl_down(bi, off, 32);
            if (ov > bv || (ov == bv && oi < bi)) { bv = ov; bi = oi; }
        }
        const int wave = tid >> 5;
        if ((tid & 31) == 0) { redV[wave] = bv; redI[wave] = bi; }
        __syncthreads();
        if (tid < 32) {
            bv = redV[tid]; bi = redI[tid];
#pragma unroll
            for (int off = 16; off > 0; off >>= 1) {
                const float ov = __shfl_down(bv, off, 32);
                const int   oi = __shfl_down(bi, off, 32);
                if (ov > bv || (ov == bv && oi < bi)) { bv = ov; bi = oi; }
            }
            if (tid == 0) {
                *sLast = bi;
                ctr[it * 3 + 0] = sX[bi];
                ctr[it * 3 + 1] = sY[bi];
                ctr[it * 3 + 2] = sZ[bi];
            }
        }
        __syncthreads();
        last = *sLast;
    }
}

// ---------------------------------------------------------------------------
// Kernel 2: KNN (top-32 smallest d2) + gather + re-center.
// Grid = BATCH*64 workgroups; each handles (batch, tile of 16 centers).
// Distances via V_WMMA_F32_16X16X4_F32 with the -2 scale and |c|^2 folded
// into the A matrix:  A = {-2cx,-2cy,-2cz,|c|^2},  B = {px,py,pz,1}
//   => acc = -2*(c.p) + |c|^2,  d2 = acc + |p|^2  (one v_add per row).
// ---------------------------------------------------------------------------
#define CHUNK   1024
#define D2S     1028                 // padded row stride for the d2 LDS tile

__global__ __launch_bounds__(256) void knn_kernel(const float* __restrict__ xyz,
                                                  float* __restrict__ out) {
    extern __shared__ char smem_raw[];
    float* sP4    = (float*)smem_raw;              // CHUNK*4 padded points
    float* sP2    = sP4 + CHUNK * 4;               // CHUNK
    float* sD2    = sP2 + CHUNK;                   // 16 * D2S
    float* sCandV = sD2 + 16 * D2S;                // 32 * 256 (slot-major)
    int*   sCandI = (int*)(sCandV + 32 * 256);     // 32 * 256
    float* sMergeV = (float*)(sCandI + 32 * 256);  // 256
    int*   sMergeI = (int*)(sMergeV + 256);        // 256
    int*   sMergeS = sMergeI + 256;                // 256
    int*   sFinal  = sMergeS + 256;                // 16 * 32
    float* sC4 = (float*)(sFinal + 16 * 32);       // 16 * {-2cx,-2cy,-2cz,|c|^2}

    const int tid = threadIdx.x;
    const int b   = blockIdx.x >> 6;
    const int ct  = blockIdx.x & 63;               // center tile (16 centers)
    const float* pts = xyz + (size_t)b * NPTS * 3;
    const float* ctr = out + CENTER_OFF + ((size_t)b * NGRP + (size_t)ct * 16) * 3;

    if (tid < 16) {
        const float cx = ctr[tid * 3 + 0];
        const float cy = ctr[tid * 3 + 1];
        const float cz = ctr[tid * 3 + 2];
        sC4[tid * 4 + 0] = -2.0f * cx;
        sC4[tid * 4 + 1] = -2.0f * cy;
        sC4[tid * 4 + 2] = -2.0f * cz;
        sC4[tid * 4 + 3] = fmaf(cx, cx, fmaf(cy, cy, cz * cz));
    }
    __syncthreads();

    const int lane = tid & 31;
    const int hi   = lane >> 4;        // 0: K=0,1 half; 1: K=2,3 half
    const int ml   = lane & 15;
    const int wave = tid >> 5;

    // A fragment: 16x4 f32 ({-2c, |c|^2}) — branchless b64 load
    const v2f afrag = *(const v2f*)(sC4 + 4 * ml + 2 * hi);

    const int cSel = tid >> 4;   // selection: center owned by this thread
    const int sub  = tid & 15;

    int   cnt = 0;
    float worstV = -1e30f;
    int   worstS = 0;

    for (int chunk = 0; chunk < NPTS / CHUNK; ++chunk) {
        // Load + pad(1.0) + |p|^2 in one pass (12B/thread contiguous reads)
        const float* src = pts + (size_t)chunk * CHUNK * 3;
        for (int i = tid; i < CHUNK; i += 256) {
            const float x = src[3 * i + 0];
            const float y = src[3 * i + 1];
            const float z = src[3 * i + 2];
            sP4[4 * i + 0] = x;
            sP4[4 * i + 1] = y;
            sP4[4 * i + 2] = z;
            sP4[4 * i + 3] = 1.0f;          // K=3 slot multiplies |c|^2
            sP2[i] = fmaf(x, x, fmaf(y, y, z * z));
        }
        __syncthreads();

        // Prefetch next chunk while we compute (global_prefetch_b8)
        if (chunk + 1 < NPTS / CHUNK) {
            const char* nxt = (const char*)(pts + (size_t)(chunk + 1) * CHUNK * 3);
            __builtin_prefetch(nxt + (size_t)tid * 48, 0, 0);
        }

        // --- WMMA phase: 64 tiles of 16 points; wave w owns 8 tiles ---
        for (int t = 0; t < 8; ++t) {
            const int n0 = (wave * 8 + t) * 16;
            const int p  = n0 + ml;
            // B fragment: one ds_load_b64 ({x,y} low half, {z,1} high half)
            const v2f bfrag = *(const v2f*)(sP4 + 4 * p + 2 * hi);
            v8f acc = {};
            acc = __builtin_amdgcn_wmma_f32_16x16x4_f32(
                false, afrag, false, bfrag, (short)0, acc, false, false);
            const float p2n = sP2[p];
#pragma unroll
            for (int r = 0; r < 8; ++r) {
                // acc = -2*(c.p) + |c|^2  =>  d2 = acc + |p|^2
                sD2[(r + 8 * hi) * D2S + p] = acc[r] + p2n;
            }
        }
        __syncthreads();

        // --- Selection phase: 16 threads per center, strided scan ---
        const float* row = sD2 + cSel * D2S;
        for (int j = 0; j < CHUNK / 16; ++j) {
            const int n = sub + j * 16;
            const float v = row[n];
            const int gidx = chunk * CHUNK + n;
            if (cnt < KNN_K) {
                sCandV[cnt * 256 + tid] = v;
                sCandI[cnt * 256 + tid] = gidx;
                if (v > worstV) { worstV = v; worstS = cnt; }
                ++cnt;
            } else if (v < worstV) {
                sCandV[worstS * 256 + tid] = v;
                sCandI[worstS * 256 + tid] = gidx;
                worstV = -1e30f;
#pragma unroll 4
                for (int s = 0; s < KNN_K; ++s) {
                    const float cv = sCandV[s * 256 + tid];
                    if (cv > worstV) { worstV = cv; worstS = s; }
                }
            }
        }
        __syncthreads();
    }

    // --- Merge: 32 rounds of cooperative argmin over 16x32 candidates ---
    for (int r = 0; r < KNN_K; ++r) {
        float mv = 1e30f;
        int   mi = 0x7fffffff;
        int   ms = 0;
#pragma unroll 4
        for (int s = 0; s < KNN_K; ++s) {
            const float v  = sCandV[s * 256 + tid];
            const int   ix = sCandI[s * 256 + tid];
            if (v < mv || (v == mv && ix < mi)) { mv = v; mi = ix; ms = s; }
        }
        sMergeV[tid] = mv; sMergeI[tid] = mi; sMergeS[tid] = ms;
        __syncthreads();
        if (sub == 0) {
            float bv = 1e30f;
            int   bIdx = 0x7fffffff;
            int   bt = tid;
            for (int u = 0; u < 16; ++u) {
                const float v  = sMergeV[tid + u];
                const int   ix = sMergeI[tid + u];
                if (v < bv || (v == bv && ix < bIdx)) { bv = v; bIdx = ix; bt = tid + u; }
            }
            sFinal[cSel * KNN_K + r] = bIdx;
            const int ws = sMergeS[bt];
            sCandV[ws * 256 + bt] = 1e30f;       // remove winner
            sCandI[ws * 256 + bt] = 0x7fffffff;
        }
        __syncthreads();
    }

    // --- Gather neighborhoods, re-center, write output ---
    // center coordinate cx == -0.5f * sC4[c*4+i]  (exact: *0.5 is exponent op)
    const int g0 = ct * 16;
    for (int o = tid; o < 16 * KNN_K; o += 256) {
        const int c = o >> 5;
        const int k = o & 31;
        const int idx = sFinal[c * KNN_K + k];
        const float x = pts[3 * idx + 0];
        const float y = pts[3 * idx + 1];
        const float z = pts[3 * idx + 2];
        const size_t base = (((size_t)b * NGRP + g0 + c) * KNN_K + k) * 3;
        out[base + 0] = fmaf(0.5f, sC4[c * 4 + 0], x);   // x - cx
        out[base + 1] = fmaf(0.5f, sC4[c * 4 + 1], y);   // y - cy
        out[base + 2] = fmaf(0.5f, sC4[c * 4 + 2], z);   // z - cz
    }
}

// ---------------------------------------------------------------------------
extern "C" void kernel_launch(void* const* d_in, const int* in_sizes, int n_in,
                              void* d_out, int out_size, void* d_ws, size_t ws_size,
                              hipStream_t stream) {
    const float* xyz = (const float*)d_in[0];   // [8,16384,3] f32
    float* out = (float*)d_out;                 // neighborhood (786432) ++ center (24576)
    (void)in_sizes; (void)n_in; (void)out_size; (void)d_ws; (void)ws_size;

    const size_t fps_smem = (size_t)(3 * NPTS + 32) * sizeof(float) + 33 * sizeof(int);
    fps_kernel<<<BATCH, 1024, fps_smem, stream>>>(xyz, out);

    const size_t knn_smem =
        (size_t)(CHUNK * 4 + CHUNK + 16 * D2S + 32 * 256) * sizeof(float) +  // sP4,sP2,sD2,sCandV
        (size_t)(32 * 256) * sizeof(int) +                                   // sCandI
        256 * sizeof(float) + 2 * 256 * sizeof(int) +                        // merge V/I/S
        16 * KNN_K * sizeof(int) +                                           // sFinal
        64 * sizeof(float);                                                  // sC4
    knn_kernel<<<BATCH * 64, 256, knn_smem, stream>>>(xyz, out);
}